// GnnBlock_29506425323528
// MI455X (gfx1250) — compile-verified
//
#include <hip/hip_runtime.h>
#include <hip/hip_bf16.h>

// Problem constants (match reference)
#define NN 50000
#define EE 625000
#define FF 128
#define NEG_SLOPE 0.01f

typedef __attribute__((ext_vector_type(2))) float v2f;
typedef __attribute__((ext_vector_type(8))) float v8f;

// ---------------------------------------------------------------------------
// Counting-sort machinery: sort edge list by destination so aggregation
// becomes a segmented gather-reduce with NO floating-point atomics.
// ---------------------------------------------------------------------------
__global__ void zero_int_kernel(int* __restrict__ p, int n) {
    int i = blockIdx.x * blockDim.x + threadIdx.x;
    if (i < n) p[i] = 0;
}

__global__ void count_kernel(const int* __restrict__ dst, int* __restrict__ count, int e_cnt) {
    int e = blockIdx.x * blockDim.x + threadIdx.x;
    if (e < e_cnt) atomicAdd(&count[dst[e]], 1);
}

// Single-block exclusive scan over N bins (LDS Hillis-Steele per 1024 chunk,
// serial carry across chunks). N=50000 -> 49 chunks; negligible runtime.
__global__ __launch_bounds__(1024)
void scan_kernel(const int* __restrict__ count, int* __restrict__ start, int n) {
    __shared__ int smem[1024];
    __shared__ int carry_s;
    if (threadIdx.x == 0) carry_s = 0;
    __syncthreads();
    for (int base = 0; base < n; base += 1024) {
        int i = base + (int)threadIdx.x;
        int v = (i < n) ? count[i] : 0;
        smem[threadIdx.x] = v;
        __syncthreads();
        for (int off = 1; off < 1024; off <<= 1) {
            int t = (threadIdx.x >= (unsigned)off) ? smem[threadIdx.x - off] : 0;
            __syncthreads();
            smem[threadIdx.x] += t;
            __syncthreads();
        }
        int incl  = smem[threadIdx.x];
        int total = smem[1023];
        int c     = carry_s;
        __syncthreads();                      // all reads done before carry update
        if (i < n) start[i] = c + incl - v;   // exclusive prefix
        if (threadIdx.x == 0) carry_s = c + total;
        __syncthreads();
    }
    if (threadIdx.x == 0) start[n] = carry_s; // == E
}

__global__ void copy_int_kernel(const int* __restrict__ a, int* __restrict__ b, int n) {
    int i = blockIdx.x * blockDim.x + threadIdx.x;
    if (i < n) b[i] = a[i];
}

__global__ void fill_kernel(const int* __restrict__ src, const int* __restrict__ dst,
                            int* __restrict__ cur, int* __restrict__ sorted_src, int e_cnt) {
    int e = blockIdx.x * blockDim.x + threadIdx.x;
    if (e < e_cnt) {
        int p = atomicAdd(&cur[dst[e]], 1);
        sorted_src[p] = src[e];
    }
}

// ---------------------------------------------------------------------------
// Segmented gather-reduce, one wave per node. Lane L owns features
// [4L, 4L+4). Each loop iteration reads one full 512B row of `feat`
// (32 lanes x float4, fully coalesced). No atomics.
// ---------------------------------------------------------------------------

// SAGE aggregate: mean_j x_j  ->  meanbuf[node]
__global__ __launch_bounds__(256)
void agg_mean_kernel(const float* __restrict__ feat,
                     const int* __restrict__ sorted_src,
                     const int* __restrict__ start,
                     float* __restrict__ meanbuf) {
    int node = blockIdx.x * 8 + (threadIdx.x >> 5);
    int lane = threadIdx.x & 31;
    if (node >= NN) return;
    int s = start[node], e = start[node + 1];
    float ax = 0.f, ay = 0.f, az = 0.f, aw = 0.f;
    for (int j = s; j < e; ++j) {
        const float4 v = *(const float4*)(feat + (long)sorted_src[j] * FF + lane * 4);
        ax += v.x; ay += v.y; az += v.z; aw += v.w;
    }
    const float ic = 1.0f / fmaxf((float)(e - s), 1.0f);
    float4 r; r.x = ax * ic; r.y = ay * ic; r.z = az * ic; r.w = aw * ic;
    *(float4*)(meanbuf + (long)node * FF + lane * 4) = r;
}

// GCN aggregate fused with epilogue: out = leaky(sum_j g_j + b_gcn) + x
__global__ __launch_bounds__(256)
void agg_final_kernel(const float* __restrict__ g,
                      const int* __restrict__ sorted_src,
                      const int* __restrict__ start,
                      const float* __restrict__ bg,
                      const float* __restrict__ x,
                      float* __restrict__ out) {
    int node = blockIdx.x * 8 + (threadIdx.x >> 5);
    int lane = threadIdx.x & 31;
    if (node >= NN) return;
    int s = start[node], e = start[node + 1];
    float ax = 0.f, ay = 0.f, az = 0.f, aw = 0.f;
    for (int j = s; j < e; ++j) {
        const float4 v = *(const float4*)(g + (long)sorted_src[j] * FF + lane * 4);
        ax += v.x; ay += v.y; az += v.z; aw += v.w;
    }
    const int f = lane * 4;
    const float4 bb = *(const float4*)(bg + f);
    const float4 xr = *(const float4*)(x + (long)node * FF + f);
    float4 r;
    float t;
    t = ax + bb.x; r.x = ((t > 0.f) ? t : t * NEG_SLOPE) + xr.x;
    t = ay + bb.y; r.y = ((t > 0.f) ? t : t * NEG_SLOPE) + xr.y;
    t = az + bb.z; r.z = ((t > 0.f) ? t : t * NEG_SLOPE) + xr.z;
    t = aw + bb.w; r.w = ((t > 0.f) ? t : t * NEG_SLOPE) + xr.w;
    *(float4*)(out + (long)node * FF + f) = r;
}

// ---------------------------------------------------------------------------
// fp32 WMMA GEMMs (V_WMMA_F32_16X16X4_F32), one wave per 16x16 output tile.
// Operand layout per ISA 7.12.2 (half = lane>>4, l15 = lane&15):
//   A: a.x = A[M=l15][K=k+2*half], a.y = A[M=l15][K=k+2*half+1]
//   B: b.x = B[K=k+2*half][N=l15], b.y = B[K=k+2*half+1][N=l15]
//   C/D: vgpr i -> (M = i + 8*half, N = l15)
// Total tiles = (50000/16)*(128/16) = 25000 -> 3125 blocks of 8 waves.
// ---------------------------------------------------------------------------

// SAGE: h = leaky( mean @ Wl + x @ Wr + b )
__global__ __launch_bounds__(256)
void sage_gemm_kernel(const float* __restrict__ x,
                      const float* __restrict__ meanbuf,
                      const float* __restrict__ wl,   // [F,F] row-major: wl[k*F+n]
                      const float* __restrict__ bl,   // [F]
                      const float* __restrict__ wr,   // [F,F]
                      float* __restrict__ h) {
    const int lane = threadIdx.x & 31;
    const int wave = threadIdx.x >> 5;
    const int tile = blockIdx.x * 8 + wave;       // [0, 25000)
    const int tn   = tile & 7;
    const int tm   = tile >> 3;
    const int half = lane >> 4;
    const int l15  = lane & 15;
    const int row  = tm * 16 + l15;
    const int col  = tn * 16 + l15;

    v8f c = {0.f, 0.f, 0.f, 0.f, 0.f, 0.f, 0.f, 0.f};
    const float* mrow = meanbuf + (long)row * FF;
    const float* xrow = x       + (long)row * FF;

    #pragma unroll 4
    for (int k = 0; k < FF; k += 4) {
        const int ka = k + half * 2;
        v2f a, b;
        a.x = mrow[ka];
        a.y = mrow[ka + 1];
        b.x = wl[ka * FF + col];
        b.y = wl[(ka + 1) * FF + col];
        c = __builtin_amdgcn_wmma_f32_16x16x4_f32(false, a, false, b,
                                                  (short)0, c, false, false);
    }
    #pragma unroll 4
    for (int k = 0; k < FF; k += 4) {
        const int ka = k + half * 2;
        v2f a, b;
        a.x = xrow[ka];
        a.y = xrow[ka + 1];
        b.x = wr[ka * FF + col];
        b.y = wr[(ka + 1) * FF + col];
        c = __builtin_amdgcn_wmma_f32_16x16x4_f32(false, a, false, b,
                                                  (short)0, c, false, false);
    }

    const float bias = bl[col];
    #pragma unroll
    for (int i = 0; i < 8; ++i) {
        int m = tm * 16 + i + half * 8;
        float v = c[i] + bias;
        v = (v > 0.0f) ? v : v * NEG_SLOPE;
        h[(long)m * FF + col] = v;
    }
}

// GCN GEMM: g = h @ Wg (bias/activation fused into agg_final)
__global__ __launch_bounds__(256)
void gcn_gemm_kernel(const float* __restrict__ hmat,
                     const float* __restrict__ wg,
                     float* __restrict__ g) {
    const int lane = threadIdx.x & 31;
    const int wave = threadIdx.x >> 5;
    const int tile = blockIdx.x * 8 + wave;
    const int tn   = tile & 7;
    const int tm   = tile >> 3;
    const int half = lane >> 4;
    const int l15  = lane & 15;
    const int row  = tm * 16 + l15;
    const int col  = tn * 16 + l15;

    v8f c = {0.f, 0.f, 0.f, 0.f, 0.f, 0.f, 0.f, 0.f};
    const float* hrow = hmat + (long)row * FF;

    #pragma unroll 4
    for (int k = 0; k < FF; k += 4) {
        const int ka = k + half * 2;
        v2f a, b;
        a.x = hrow[ka];
        a.y = hrow[ka + 1];
        b.x = wg[ka * FF + col];
        b.y = wg[(ka + 1) * FF + col];
        c = __builtin_amdgcn_wmma_f32_16x16x4_f32(false, a, false, b,
                                                  (short)0, c, false, false);
    }
    #pragma unroll
    for (int i = 0; i < 8; ++i) {
        int m = tm * 16 + i + half * 8;
        g[(long)m * FF + col] = c[i];
    }
}

// ---------------------------------------------------------------------------
extern "C" void kernel_launch(void* const* d_in, const int* in_sizes, int n_in,
                              void* d_out, int out_size, void* d_ws, size_t ws_size,
                              hipStream_t stream) {
    const float* x  = (const float*)d_in[0];
    const int*   ei = (const int*)  d_in[1];   // [2,E]: row0=src, row1=dst
    const float* wl = (const float*)d_in[2];
    const float* bl = (const float*)d_in[3];
    const float* wr = (const float*)d_in[4];
    const float* wg = (const float*)d_in[5];
    const float* bg = (const float*)d_in[6];
    float* out = (float*)d_out;

    const int* src = ei;
    const int* dst = ei + EE;

    const long NF = (long)NN * FF;

    // Workspace: floats [meanbuf NF | hbuf NF | gbuf NF] then ints
    // [count N | start N+1 | cur N | sorted_src E]
    float* fws     = (float*)d_ws;
    float* meanbuf = fws;
    float* hbuf    = fws + NF;
    float* gbuf    = fws + 2 * NF;
    int*   iws     = (int*)(fws + 3 * NF);
    int*   count   = iws;
    int*   start   = count + NN;
    int*   cur     = start + NN + 1;
    int*   ssrc    = cur + NN;

    const int T = 256;
    const int gemm_blocks = (NN / 16) * (FF / 16) / 8;   // 3125
    const int node_blocks = (NN + 7) / 8;                // 6250 (wave per node)
    const int edge_blocks = (EE + T - 1) / T;

    // --- build CSR (sort edges by dst) ---
    zero_int_kernel<<<(NN + T - 1) / T, T, 0, stream>>>(count, NN);
    count_kernel<<<edge_blocks, T, 0, stream>>>(dst, count, EE);
    scan_kernel<<<1, 1024, 0, stream>>>(count, start, NN);
    copy_int_kernel<<<(NN + T - 1) / T, T, 0, stream>>>(start, cur, NN);
    fill_kernel<<<edge_blocks, T, 0, stream>>>(src, dst, cur, ssrc, EE);

    // --- SAGE ---
    agg_mean_kernel<<<node_blocks, T, 0, stream>>>(x, ssrc, start, meanbuf);
    sage_gemm_kernel<<<gemm_blocks, T, 0, stream>>>(x, meanbuf, wl, bl, wr, hbuf);

    // --- GCN ---
    gcn_gemm_kernel<<<gemm_blocks, T, 0, stream>>>(hbuf, wg, gbuf);
    agg_final_kernel<<<node_blocks, T, 0, stream>>>(gbuf, ssrc, start, bg, x, out);
}